// AdaptiveDepthWiseConv2d_72868415144241
// MI455X (gfx1250) — compile-verified
//
#include <hip/hip_runtime.h>

#define BB 32
#define CC 256
#define HH 56
#define WW 56
#define KK 3
#define HW (HH * WW)      // 3136
#define NCOL (CC * KK)    // 768
#define TW 58             // halo tile width

typedef __attribute__((ext_vector_type(2))) float v2f;
typedef __attribute__((ext_vector_type(8))) float v8f;

// -------- Kernel 1: xm[b][c] = mean over HxW of x[b][c] ------------------
// One block per (b,c) plane; float4 loads (plane is 16B aligned: 3136*4 bytes).
// Default (RT) cache policy: this streams all of x through L2 (103MB < 192MB),
// priming it for the conv kernel's second read.
__global__ __launch_bounds__(256) void mean_kernel(const float* __restrict__ x,
                                                   float* __restrict__ xm) {
    const int plane = blockIdx.x;  // b*C + c
    const float4* p = (const float4*)(x + (size_t)plane * HW);
    float s = 0.0f;
    for (int i = threadIdx.x; i < HW / 4; i += 256) {
        float4 v = p[i];
        s += v.x + v.y + v.z + v.w;
    }
    __shared__ float red[256];
    red[threadIdx.x] = s;
    __syncthreads();
    for (int off = 128; off > 0; off >>= 1) {
        if (threadIdx.x < (unsigned)off) red[threadIdx.x] += red[threadIdx.x + off];
        __syncthreads();
    }
    if (threadIdx.x == 0) xm[plane] = red[0] * (1.0f / HW);
}

// -------- Kernel 2a: logits = xm (32x256) @ w1^T (256x768) via WMMA f32 ---
// One wave32 per 16x16 output tile; V_WMMA_F32_16X16X4_F32, 64 k-steps.
// A (16x4 f32): lane l<16 -> {A[m][k0],A[m][k0+1]}, l>=16 -> {A[m][k0+2],A[m][k0+3]}
// B (4x16 f32): B[k][n] = w1[n][k]; same split across lane halves.
__global__ __launch_bounds__(32) void gemm_kernel(const float* __restrict__ xm,
                                                  const float* __restrict__ w1,
                                                  float* __restrict__ logits) {
    const int n0   = blockIdx.x * 16;
    const int m0   = blockIdx.y * 16;
    const int lane = threadIdx.x;
    const int lm   = lane & 15;
    const int half = lane >> 4;

    const float* arow = xm + (size_t)(m0 + lm) * CC;  // A row m0+lm
    const float* brow = w1 + (size_t)(n0 + lm) * CC;  // B column n0+lm (w1 row)

    v8f acc = {};
    for (int k0 = 0; k0 < CC; k0 += 4) {
        v2f a, b;
        a.x = arow[k0 + 2 * half];
        a.y = arow[k0 + 2 * half + 1];
        b.x = brow[k0 + 2 * half];
        b.y = brow[k0 + 2 * half + 1];
        acc = __builtin_amdgcn_wmma_f32_16x16x4_f32(
            /*neg_a=*/false, a, /*neg_b=*/false, b,
            /*c_mod=*/(short)0, acc, /*reuse_a=*/false, /*reuse_b=*/false);
    }
    // D layout: VGPR r, lanes 0-15 -> D[m0+r][n0+lm]; lanes 16-31 -> D[m0+8+r][n0+lm]
    const int m = m0 + 8 * half;
    const int n = n0 + lm;
#pragma unroll
    for (int r = 0; r < 8; ++r)
        logits[(size_t)(m + r) * NCOL + n] = acc[r];
}

// -------- Kernel 2b: softmax over K=3 + collapse K depthwise kernels ------
// w_eff[b,c][0..8] = sum_j prob[b,c,j] * cw[j][c][0..8]
__global__ __launch_bounds__(256) void gate_kernel(const float* __restrict__ logits,
                                                   const float* __restrict__ cw,
                                                   float* __restrict__ weff) {
    const int idx = blockIdx.x * 256 + threadIdx.x;  // b*C + c
    if (idx >= BB * CC) return;
    const int b = idx / CC;
    const int c = idx % CC;

    const float* l = logits + (size_t)b * NCOL + c * KK;
    float l0 = l[0], l1 = l[1], l2 = l[2];
    float mx = fmaxf(l0, fmaxf(l1, l2));
    float e0 = expf(l0 - mx), e1 = expf(l1 - mx), e2 = expf(l2 - mx);
    float inv = 1.0f / (e0 + e1 + e2);
    float p0 = e0 * inv, p1 = e1 * inv, p2 = e2 * inv;

    const float* c0 = cw + (size_t)0 * CC * 9 + (size_t)c * 9;
    const float* c1 = cw + (size_t)1 * CC * 9 + (size_t)c * 9;
    const float* c2 = cw + (size_t)2 * CC * 9 + (size_t)c * 9;
    float* o = weff + (size_t)idx * 9;
#pragma unroll
    for (int t = 0; t < 9; ++t)
        o[t] = p0 * c0[t] + p1 * c1[t] + p2 * c2[t];
}

// -------- Kernel 3: depthwise 3x3 conv with per-plane effective kernel ----
// One block per (b,c) plane. 58x58 halo tile in LDS (13.5 KB / 320 KB WGP).
// x re-read hits L2 (primed by kernel 1); output uses NT stores so the
// streaming writes don't evict x from L2.
__global__ __launch_bounds__(256) void dwconv_kernel(const float* __restrict__ x,
                                                     const float* __restrict__ weff,
                                                     float* __restrict__ out) {
    const int plane = blockIdx.x;  // b*C + c
    __shared__ float tile[TW * TW];
    const float* xin = x + (size_t)plane * HW;

    for (int i = threadIdx.x; i < TW * TW; i += 256) {
        int r = i / TW, cc = i % TW;
        int gr = r - 1, gc = cc - 1;
        float v = 0.0f;
        if (gr >= 0 && gr < HH && gc >= 0 && gc < WW) v = xin[gr * WW + gc];
        tile[i] = v;
    }

    const float* wv = weff + (size_t)plane * 9;
    float w0 = wv[0], w1 = wv[1], w2 = wv[2];
    float w3 = wv[3], w4 = wv[4], w5 = wv[5];
    float w6 = wv[6], w7 = wv[7], w8 = wv[8];
    __syncthreads();

    float* o = out + (size_t)plane * HW;
    for (int i = threadIdx.x; i < HW; i += 256) {
        int r = i / WW, cc = i % WW;
        const float* t0 = &tile[r * TW + cc];
        float s = t0[0]        * w0 + t0[1]          * w1 + t0[2]          * w2
                + t0[TW]       * w3 + t0[TW + 1]     * w4 + t0[TW + 2]     * w5
                + t0[2 * TW]   * w6 + t0[2 * TW + 1] * w7 + t0[2 * TW + 2] * w8;
        __builtin_nontemporal_store(s, &o[i]);
    }
}

extern "C" void kernel_launch(void* const* d_in, const int* in_sizes, int n_in,
                              void* d_out, int out_size, void* d_ws, size_t ws_size,
                              hipStream_t stream) {
    const float* x  = (const float*)d_in[0];  // (B, C, H, W)
    const float* cw = (const float*)d_in[1];  // (K, C, 1, 3, 3)
    const float* w1 = (const float*)d_in[2];  // (C*K, C)
    float* out = (float*)d_out;               // (B, C, H, W)

    float* xm     = (float*)d_ws;             // B*C          = 8192 f32
    float* logits = xm + BB * CC;             // B*C*K        = 24576 f32
    float* weff   = logits + BB * NCOL;       // B*C*9        = 73728 f32

    mean_kernel<<<BB * CC, 256, 0, stream>>>(x, xm);
    gemm_kernel<<<dim3(NCOL / 16, BB / 16), 32, 0, stream>>>(xm, w1, logits);
    gate_kernel<<<(BB * CC) / 256, 256, 0, stream>>>(logits, cw, weff);
    dwconv_kernel<<<BB * CC, 256, 0, stream>>>(x, weff, out);
}